// MeanShift_loss_35261681500647
// MI455X (gfx1250) — compile-verified
//
#include <hip/hip_runtime.h>
#include <hip/hip_bf16.h>
#include <math.h>

typedef __attribute__((ext_vector_type(2))) float v2f;
typedef __attribute__((ext_vector_type(8))) float v8f;

// ws[0] = sum of row-mins pass 1, ws[1] = sum of row-mins pass 2
__global__ void chamfer_init_kernel(float* ws) {
    if (threadIdx.x == 0 && blockIdx.x == 0) { ws[0] = 0.0f; ws[1] = 0.0f; }
}

// For each row point x in X[b], compute min_y ||x - y|| over all Y[b] via
//   d2 = (||x||^2 - 2 x.y) + ||y||^2
// using V_WMMA_F32_16X16X4_F32 per 16x16 tile:
//   A row m  = (x0, x1, x2, ||x||^2)      (K = 4)
//   B col n  = (-2y0, -2y1, -2y2, 1)
//   C[v]     = ||y_n||^2  (N depends only on lane -> broadcast across VGPRs)
// so the WMMA result IS the squared-distance tile. Running min kept on d2
// (sqrt is monotone); sqrt applied once per row at the end.
//
// Each wave owns 16 rows and sweeps all column tiles. Block = 256 thr = 8
// waves = 128 rows. Row mins are reduced with intra-half shfl_xor (offsets
// 8,4,2,1 never cross the 16-lane halves), summed into *sum_acc, and
// optionally written to row_mins_out (pass 2 -> mins_seeds output).
template <bool WRITE_MINS>
__global__ __launch_bounds__(256)
void chamfer_rowmin_kernel(const float* __restrict__ X,
                           const float* __restrict__ Y,
                           int rowsPerBatch, int colsPerBatch,
                           float* __restrict__ sum_acc,
                           float* __restrict__ row_mins_out) {
    const int blocksPerBatch = rowsPerBatch / 128;
    const int b    = blockIdx.x / blocksPerBatch;
    const int rb   = blockIdx.x % blocksPerBatch;
    const int wave = threadIdx.x >> 5;   // wave32
    const int lane = threadIdx.x & 31;
    const int hi   = lane >> 4;          // 0: holds K=0,1   1: holds K=2,3
    const int nrow = lane & 15;          // row index within tile (A), col (B/C)
    const int r0   = rb * 128 + wave * 16;
    const int m    = r0 + nrow;

    // Build the wave-invariant A operand (16x4 f32 -> 2 VGPRs/lane).
    const float* xp = X + ((size_t)b * rowsPerBatch + m) * 3;
    const float x0 = xp[0], x1 = xp[1], x2 = xp[2];
    const float xn = x0 * x0 + x1 * x1 + x2 * x2;
    v2f a;
    a[0] = hi ? x2 : x0;
    a[1] = hi ? xn : x1;

    v8f runmin;
#pragma unroll
    for (int v = 0; v < 8; ++v) runmin[v] = 3.4e38f;

    const float* Yb = Y + (size_t)b * colsPerBatch * 3;
    for (int t = 0; t < colsPerBatch; t += 16) {
        const int n = t + nrow;
        const float* yp = Yb + (size_t)n * 3;
        const float y0 = yp[0], y1 = yp[1], y2 = yp[2];
        const float yn = y0 * y0 + y1 * y1 + y2 * y2;
        v2f bm;
        bm[0] = hi ? (-2.0f * y2) : (-2.0f * y0);
        bm[1] = hi ? 1.0f         : (-2.0f * y1);
        v8f c;
#pragma unroll
        for (int v = 0; v < 8; ++v) c[v] = yn;   // + ||y||^2 folded into C
        // d2[v][lane]: M = v + 8*(lane>=16) + r0, N = t + (lane&15)
        v8f d2 = __builtin_amdgcn_wmma_f32_16x16x4_f32(
            false, a, false, bm, (short)0, c, false, false);
#pragma unroll
        for (int v = 0; v < 8; ++v) runmin[v] = fminf(runmin[v], d2[v]);
    }

    // Min-reduce across the 16 lanes of each half (rows v / v+8 live in
    // lanes 0-15 / 16-31 of VGPR v). Offsets 8..1 stay within each half.
#pragma unroll
    for (int off = 8; off >= 1; off >>= 1) {
#pragma unroll
        for (int v = 0; v < 8; ++v)
            runmin[v] = fminf(runmin[v], __shfl_xor(runmin[v], off, 32));
    }

    // Lane 0 owns rows r0+0..7, lane 16 owns rows r0+8..15.
    if (lane == 0 || lane == 16) {
        const int mbase = r0 + (hi ? 8 : 0);
        float s = 0.0f;
#pragma unroll
        for (int v = 0; v < 8; ++v) {
            float d = sqrtf(fmaxf(runmin[v], 0.0f));
            s += d;
            if (WRITE_MINS)
                row_mins_out[(size_t)b * rowsPerBatch + mbase + v] = d;
        }
        atomicAdd(sum_acc, s);
    }
}

__global__ void chamfer_finalize_kernel(const float* __restrict__ ws,
                                        float* __restrict__ out,
                                        float invBN, float invBM) {
    if (threadIdx.x == 0 && blockIdx.x == 0)
        out[0] = ws[0] * invBN + ws[1] * invBM;
}

extern "C" void kernel_launch(void* const* d_in, const int* in_sizes, int n_in,
                              void* d_out, int out_size, void* d_ws, size_t ws_size,
                              hipStream_t stream) {
    const float* true_pos = (const float*)d_in[0];  // [B, N, 3] f32
    const float* pred_pos = (const float*)d_in[1];  // [B, M, 3] f32
    float* out = (float*)d_out;                     // [0]=loss, [1..B*M]=mins_seeds
    float* ws  = (float*)d_ws;

    const int B = 4;
    const int N = in_sizes[0] / (B * 3);
    const int M = in_sizes[1] / (B * 3);

    chamfer_init_kernel<<<1, 64, 0, stream>>>(ws);

    // Pass 1: rows = true points, mins over preds -> Sum into ws[0].
    chamfer_rowmin_kernel<false><<<dim3(B * (N / 128)), 256, 0, stream>>>(
        true_pos, pred_pos, N, M, ws + 0, nullptr);

    // Pass 2: rows = pred points, mins over trues -> mins_seeds + Sum ws[1].
    chamfer_rowmin_kernel<true><<<dim3(B * (M / 128)), 256, 0, stream>>>(
        pred_pos, true_pos, M, N, ws + 1, out + 1);

    chamfer_finalize_kernel<<<1, 64, 0, stream>>>(
        ws, out, 1.0f / ((float)B * N), 1.0f / ((float)B * M));
}